// SCCNNLayer_44117904065323
// MI455X (gfx1250) — compile-verified
//
#include <hip/hip_runtime.h>
#include <hip/hip_bf16.h>

typedef __attribute__((ext_vector_type(2))) float v2f;
typedef __attribute__((ext_vector_type(8))) float v8f;

#define CDIM 16
#define UNROLL 8   // WMMA steps per load group; K advance per group = 32

// finite check: NaN fails any comparison, |inf| > FLT_MAX
__device__ __forceinline__ float clean_f(float v) {
    return (__builtin_fabsf(v) <= 3.402823466e+38f) ? v : 0.0f;
}

// -------- rowsum -> inverse (aggr_norm denominator) ------------------------
__global__ void __launch_bounds__(256)
rowinv_kernel(const float* __restrict__ L, float* __restrict__ inv, int N) {
    __shared__ float red[256];
    const int row = blockIdx.x;
    const float* Lr = L + (size_t)row * N;
    float s = 0.0f;
    for (int c = threadIdx.x; c < N; c += 256) s += Lr[c];
    red[threadIdx.x] = s;
    __syncthreads();
    #pragma unroll
    for (int off = 128; off > 0; off >>= 1) {
        if (threadIdx.x < off) red[threadIdx.x] += red[threadIdx.x + off];
        __syncthreads();
    }
    if (threadIdx.x == 0) {
        float iv = 1.0f / red[0];
        inv[row] = clean_f(iv);
    }
}

// -------- [N,16] -> [16,N] transpose (inputs only; stages come out T) ------
__global__ void __launch_bounds__(256)
transpose16_kernel(const float* __restrict__ in, float* __restrict__ outT, int N) {
    const int idx = blockIdx.x * blockDim.x + threadIdx.x;  // idx = c*N + n
    if (idx >= N * CDIM) return;
    const int c = idx / N;
    const int n = idx - c * N;
    outT[idx] = in[n * CDIM + c];   // coalesced writes
}

// Load one pipeline group: 8 A-fragments + 8 B-fragments, all dense b64
__device__ __forceinline__ void load_group(const float* __restrict__ Arow,
                                           const float* __restrict__ Brow,
                                           int k, v2f* a, v2f* b) {
    #pragma unroll
    for (int u = 0; u < UNROLL; ++u) {
        const int kk = k + 4 * u;
        a[u] = *(const v2f*)(Arow + kk);  // A: (row=laneLo, K=kk+2*laneHi+{0,1})
        b[u] = *(const v2f*)(Brow + kk);  // B: (col=laneLo, K=kk+2*laneHi+{0,1})
    }
}

__device__ __forceinline__ void mma_group(v8f& acc, const v2f* a, const v2f* b) {
    #pragma unroll
    for (int u = 0; u < UNROLL; ++u)
        acc = __builtin_amdgcn_wmma_f32_16x16x4_f32(
            false, a[u], false, b[u], (short)0, acc, false, false);
}

// -------- YT[16,M] = (A[MxK] @ XT^T[Kx16]) stored transposed ---------------
// One 16-row tile per block; 8 wave32 split K 8 ways; 2-stage software
// pipeline; both operands are dense per-lane b64 streams on LOADcnt.
__global__ void __launch_bounds__(256)
wmma_gemm16(const float* __restrict__ A,   // M x K row-major (ld == K)
            const float* __restrict__ XT,  // 16 x K row-major (transposed X)
            float* __restrict__ YT,        // 16 x M row-major (transposed out)
            int K,                         // multiple of 512
            int M,
            const float* __restrict__ inv) // nullable: row scale + finite-clean
{
    __shared__ float lds[7][32][8];

    const int tid    = threadIdx.x;
    const int lane   = tid & 31;
    const int wave   = tid >> 5;          // K-slice id, 0..7
    const int m0     = blockIdx.x * 16;
    const int laneHi = lane >> 4;         // 0: K={0,1}, 1: K={2,3}
    const int laneLo = lane & 15;

    const int kPer = K >> 3;              // per-wave K span (multiple of 64)
    const float* Arow = A  + (size_t)(m0 + laneLo) * (size_t)K + (laneHi << 1);
    const float* Brow = XT + (size_t)laneLo * (size_t)K + (laneHi << 1);

    v8f acc = {};
    v2f a0[UNROLL], b0[UNROLL], a1[UNROLL], b1[UNROLL];

    int k = wave * kPer;
    load_group(Arow, Brow, k, a0, b0);
    const int nPairs = kPer >> 6;         // pairs of 32-wide K groups (>=1)
    for (int i = 0; i < nPairs - 1; ++i) {
        load_group(Arow, Brow, k + 32, a1, b1);
        mma_group(acc, a0, b0);
        load_group(Arow, Brow, k + 64, a0, b0);
        mma_group(acc, a1, b1);
        k += 64;
    }
    load_group(Arow, Brow, k + 32, a1, b1);
    mma_group(acc, a0, b0);
    mma_group(acc, a1, b1);

    // cross-wave K reduction through LDS
    if (wave != 0) {
        #pragma unroll
        for (int r = 0; r < 8; ++r) lds[wave - 1][lane][r] = acc[r];
    }
    __syncthreads();
    if (wave == 0) {
        #pragma unroll
        for (int s = 0; s < 7; ++s)
            #pragma unroll
            for (int r = 0; r < 8; ++r) acc[r] += lds[s][lane][r];

        // C/D layout: VGPR r, lane l -> row = r + 8*(l>=16), col = l&15
        #pragma unroll
        for (int r = 0; r < 8; ++r) {
            const int row = m0 + r + laneHi * 8;
            float v = acc[r];
            if (inv) {
                v *= inv[row];     // aggr_norm scale
                v = clean_f(v);    // where(isfinite, ., 0)
            }
            YT[laneLo * M + row] = v;   // store transposed for next stage
        }
    }
}

// -------- y0[n,o] = sum_{i,k} Xall[n,i,k] * W[i,o,k] (stages are [16,N0]) --
__global__ void __launch_bounds__(256)
finalize_kernel(const float* __restrict__ S0, const float* __restrict__ S1,
                const float* __restrict__ S2, const float* __restrict__ S3,
                const float* __restrict__ S4, const float* __restrict__ S5,
                const float* __restrict__ S6, const float* __restrict__ S7,
                const float* __restrict__ W, float* __restrict__ out, int N0) {
    const int idx = blockIdx.x * blockDim.x + threadIdx.x;
    if (idx >= N0 * CDIM) return;
    const int n = idx >> 4;
    const int o = idx & 15;
    const float* S[8] = {S0, S1, S2, S3, S4, S5, S6, S7};
    float acc = 0.0f;
    #pragma unroll
    for (int i = 0; i < CDIM; ++i) {
        const float* Wio = W + (i * CDIM + o) * 8;  // W[i,o,k], k stride 1
        #pragma unroll
        for (int kk = 0; kk < 8; ++kk)
            acc += S[kk][i * N0 + n] * Wio[kk];
    }
    out[idx] = acc;
}

extern "C" void kernel_launch(void* const* d_in, const int* in_sizes, int n_in,
                              void* d_out, int out_size, void* d_ws, size_t ws_size,
                              hipStream_t stream) {
    (void)n_in; (void)out_size; (void)ws_size;
    const float* x0 = (const float*)d_in[0];   // [N0,16]
    const float* x1 = (const float*)d_in[1];   // [N1,16]
    const float* L0 = (const float*)d_in[3];   // [N0,N0]
    const float* b1 = (const float*)d_in[8];   // [N0,N1]
    const float* W0 = (const float*)d_in[10];  // [16,16,8]

    const int N0 = in_sizes[0] / CDIM;         // 4096
    const int N1 = in_sizes[1] / CDIM;         // 8192

    float* ws   = (float*)d_ws;
    float* inv0 = ws;                 ws += N0;
    float* x0T  = ws;                 ws += N0 * CDIM;   // [16,N0]
    float* x1T  = ws;                 ws += N1 * CDIM;   // [16,N1]
    float* X1T  = ws;                 ws += N0 * CDIM;
    float* X2T  = ws;                 ws += N0 * CDIM;
    float* X3T  = ws;                 ws += N0 * CDIM;
    float* UT   = ws;                 ws += N0 * CDIM;
    float* Y1T  = ws;                 ws += N0 * CDIM;
    float* Y2T  = ws;                 ws += N0 * CDIM;
    float* Y3T  = ws;                 ws += N0 * CDIM;

    const int gemmBlocks = N0 / 16;            // one 16-row tile per block

    // inv0 = clean(1/rowsum(L0)); also warms L2 with L0 (64 MB, L2 = 192 MB)
    rowinv_kernel<<<N0, 256, 0, stream>>>(L0, inv0, N0);

    // transpose skinny inputs into the B-friendly [16,K] layout
    transpose16_kernel<<<(N0 * CDIM + 255) / 256, 256, 0, stream>>>(x0, x0T, N0);
    transpose16_kernel<<<(N1 * CDIM + 255) / 256, 256, 0, stream>>>(x1, x1T, N1);

    // Chebyshev chain on x_0 (stages produced transposed)
    wmma_gemm16<<<gemmBlocks, 256, 0, stream>>>(L0, x0T, X1T, N0, N0, inv0);
    wmma_gemm16<<<gemmBlocks, 256, 0, stream>>>(L0, X1T, X2T, N0, N0, inv0);
    wmma_gemm16<<<gemmBlocks, 256, 0, stream>>>(L0, X2T, X3T, N0, N0, inv0);

    // U = b1 @ x_1 (no norm), then Chebyshev chain on U
    wmma_gemm16<<<gemmBlocks, 256, 0, stream>>>(b1, x1T, UT, N1, N0, nullptr);
    wmma_gemm16<<<gemmBlocks, 256, 0, stream>>>(L0, UT,  Y1T, N0, N0, inv0);
    wmma_gemm16<<<gemmBlocks, 256, 0, stream>>>(L0, Y1T, Y2T, N0, N0, inv0);
    wmma_gemm16<<<gemmBlocks, 256, 0, stream>>>(L0, Y2T, Y3T, N0, N0, inv0);

    // y0 = einsum over the 8 concatenated (transposed) stages with weight_0
    finalize_kernel<<<(N0 * CDIM + 255) / 256, 256, 0, stream>>>(
        x0T, X1T, X2T, X3T, UT, Y1T, Y2T, Y3T, W0, (float*)d_out, N0);
}